// CapsuleLayer_72593537237716
// MI455X (gfx1250) — compile-verified
//
#include <hip/hip_runtime.h>
#include <hip/hip_bf16.h>
#include <math.h>

// ---- problem constants (match reference hyper-parameters) ----
#define NUM_IN   8
#define IN_DIM   16
#define KS       5
#define PADW     2
#define NUM_OUT  8
#define OUT_DIM  16
#define OD       128            // NUM_OUT*OUT_DIM
#define KDIM     400            // IN_DIM*KS*KS
#define NB       8              // batch
#define HGT      64
#define WID      64
#define HW       4096           // HGT*WID

typedef float v2f __attribute__((ext_vector_type(2)));
typedef float v8f __attribute__((ext_vector_type(8)));

// =====================================================================
// Conv as implicit GEMM with V_WMMA_F32_16X16X4_F32.
// Block = (i, n, 32-pixel row tile). 8 waves; wave w owns channels
// [16w,16w+16), two 16x16 output tiles (pixel halves). K=400 in steps of 4.
//
// Fragment layouts per CDNA5 ISA 7.12.2 (wave32):
//   A 16x4 f32 : lane l<16 -> row M=l, holds K=0,1 in v0,v1; lane l+16 holds K=2,3
//   B 4x16 f32 : lane l<16 -> col N=l, holds K=0,1 in v0,v1; lane l+16 holds K=2,3
//   C/D 16x16  : VGPR r, lanes 0-15 -> (M=r, N=lane); lanes 16-31 -> (M=r+8, N=lane-16)
// =====================================================================
__global__ __launch_bounds__(256)
void caps_conv_wmma(const float* __restrict__ u, const float* __restrict__ W,
                    float* __restrict__ uhat)
{
    __shared__ float sm[IN_DIM * KS * 36];   // input halo tile: cin x dy x (32+4) = 2880 f
    __shared__ int   ktab[KDIM];             // k -> LDS float offset of (cin,ky,kx=+0)

    const int bid = blockIdx.x;
    const int t   = bid & 127;               // pixel tile: 2 per row, 64 rows
    const int n   = (bid >> 7) & 7;
    const int i   = bid >> 10;
    const int y   = t >> 1;
    const int xb  = (t & 1) << 5;            // 0 or 32

    const int tid = threadIdx.x;

    // ---- stage input halo tile (zero-padded) ----
    for (int e = tid; e < IN_DIM * KS * 36; e += 256) {
        int cin = e / (KS * 36);
        int rem = e - cin * (KS * 36);
        int dy  = rem / 36;
        int xi  = rem - dy * 36;
        int gy  = y + dy - PADW;
        int gx  = xb + xi - PADW;
        float val = 0.0f;
        if ((unsigned)gy < HGT && (unsigned)gx < WID)
            val = u[(((size_t)n * NUM_IN + i) * IN_DIM + cin) * HW + gy * WID + gx];
        sm[e] = val;
    }
    // ---- k -> LDS offset table:  off = (cin*5+ky)*36 + kx ----
    for (int k = tid; k < KDIM; k += 256) {
        int cin = k / 25;
        int rem = k - cin * 25;
        int ky  = rem / 5;
        int kx  = rem - ky * 5;
        ktab[k] = (cin * 5 + ky) * 36 + kx;
    }
    __syncthreads();

    const int lane = tid & 31;
    const int wave = tid >> 5;        // 0..7 -> 16-channel slice
    const int l15  = lane & 15;
    const int lhi  = lane >> 4;       // 0 or 1
    const int chb  = wave * 16;

    // A-fragment pointer: W[i][chb+l15][k0], k0 starts at 2*lhi, advances by 4
    const float* wp = W + ((size_t)i * OD + chb + l15) * KDIM + 2 * lhi;

    v8f acc0 = {0.f,0.f,0.f,0.f,0.f,0.f,0.f,0.f};
    v8f acc1 = acc0;

    #pragma unroll 4
    for (int kb = 0; kb < KDIM; kb += 4) {
        v2f a = *(const v2f*)wp;          // A[l15][k0], A[l15][k0+1]
        wp += 4;
        int k0 = kb + 2 * lhi;
        int o0 = ktab[k0];
        int o1 = ktab[k0 + 1];
        v2f b0 = { sm[o0 + l15],      sm[o1 + l15]      };   // pixel cols 0..15
        v2f b1 = { sm[o0 + l15 + 16], sm[o1 + l15 + 16] };   // pixel cols 16..31
        acc0 = __builtin_amdgcn_wmma_f32_16x16x4_f32(false, a, false, b0,
                                                     (short)0, acc0, false, false);
        acc1 = __builtin_amdgcn_wmma_f32_16x16x4_f32(false, a, false, b1,
                                                     (short)0, acc1, false, false);
    }

    // ---- store D tiles: uhat[i][n][ch][pix], channel-major ----
    float* outb = uhat + ((size_t)i * NB + n) * OD * HW;
    const int pix0 = y * WID + xb;
    #pragma unroll
    for (int r = 0; r < 8; ++r) {
        int ch = chb + r + lhi * 8;
        outb[(size_t)ch * HW + pix0 + l15]      = acc0[r];
        outb[(size_t)ch * HW + pix0 + 16 + l15] = acc1[r];
    }
}

// =====================================================================
// Routing kernels (fp32, memory-bound, fully coalesced along pixels)
// Layouts: b,c/r : [I][N][O][HW]   planes t* : [I*N][HW]
//          uhat  : [I][N][OD][HW]  v (=d_out): [N][O][D][HW]
// =====================================================================
__global__ void k_zero(float* __restrict__ p, int n)
{
    int idx = blockIdx.x * blockDim.x + threadIdx.x;
    if (idx < n) p[idx] = 0.0f;
}

// t0[row][pix] = max_o b[row][o][pix]
__global__ void k_max_o(const float* __restrict__ b, float* __restrict__ t0)
{
    int idx = blockIdx.x * blockDim.x + threadIdx.x;     // [I*N*HW)
    int pix = idx & (HW - 1);
    int row = idx >> 12;
    const float* bp = b + ((size_t)row * NUM_OUT) * HW + pix;
    float m = bp[0];
    #pragma unroll
    for (int o = 1; o < NUM_OUT; ++o) m = fmaxf(m, bp[(size_t)o * HW]);
    t0[idx] = m;
}

// 5-tap horizontal max, -inf pad
__global__ void k_hmax(const float* __restrict__ s, float* __restrict__ d)
{
    int idx = blockIdx.x * blockDim.x + threadIdx.x;
    int x   = idx & (WID - 1);
    float m = -__builtin_inff();
    #pragma unroll
    for (int dx = -2; dx <= 2; ++dx) {
        int xx = x + dx;
        if ((unsigned)xx < WID) m = fmaxf(m, s[idx + dx]);
    }
    d[idx] = m;
}

// 5-tap vertical max, -inf pad
__global__ void k_vmax(const float* __restrict__ s, float* __restrict__ d)
{
    int idx = blockIdx.x * blockDim.x + threadIdx.x;
    int y   = (idx >> 6) & (HGT - 1);
    float m = -__builtin_inff();
    #pragma unroll
    for (int dy = -2; dy <= 2; ++dy) {
        int yy = y + dy;
        if ((unsigned)yy < HGT) m = fmaxf(m, s[idx + dy * WID]);
    }
    d[idx] = m;
}

// c = exp(b - bmax);  t0 = sum_o c
__global__ void k_exp(const float* __restrict__ b, const float* __restrict__ bmax,
                      float* __restrict__ c, float* __restrict__ t0)
{
    int idx = blockIdx.x * blockDim.x + threadIdx.x;     // [I*N*HW)
    int pix = idx & (HW - 1);
    int row = idx >> 12;
    size_t base = ((size_t)row * NUM_OUT) * HW + pix;
    float bm = bmax[idx];
    float s  = 0.0f;
    #pragma unroll
    for (int o = 0; o < NUM_OUT; ++o) {
        float e = expf(b[base + (size_t)o * HW] - bm);
        c[base + (size_t)o * HW] = e;
        s += e;
    }
    t0[idx] = s;
}

// 5-tap horizontal sum, 0 pad
__global__ void k_hsum(const float* __restrict__ s, float* __restrict__ d)
{
    int idx = blockIdx.x * blockDim.x + threadIdx.x;
    int x   = idx & (WID - 1);
    float a = 0.0f;
    #pragma unroll
    for (int dx = -2; dx <= 2; ++dx) {
        int xx = x + dx;
        if ((unsigned)xx < WID) a += s[idx + dx];
    }
    d[idx] = a;
}

// 5-tap vertical sum, 0 pad
__global__ void k_vsum(const float* __restrict__ s, float* __restrict__ d)
{
    int idx = blockIdx.x * blockDim.x + threadIdx.x;
    int y   = (idx >> 6) & (HGT - 1);
    float a = 0.0f;
    #pragma unroll
    for (int dy = -2; dy <= 2; ++dy) {
        int yy = y + dy;
        if ((unsigned)yy < HGT) a += s[idx + dy * WID];
    }
    d[idx] = a;
}

// r = c / sum_c   (broadcast over o)
__global__ void k_div(float* __restrict__ c, const float* __restrict__ sc)
{
    int idx = blockIdx.x * blockDim.x + threadIdx.x;     // [I*N*O*HW)
    int pix = idx & (HW - 1);
    int row = idx >> 15;                                 // /(O*HW)
    c[idx] = c[idx] / sc[row * HW + pix];
}

// p[n][od][pix] = sum_i r[i][n][od>>4][pix] * uhat[i][n][od][pix]
__global__ void k_p(const float* __restrict__ r, const float* __restrict__ uhat,
                    float* __restrict__ p)
{
    int idx = blockIdx.x * blockDim.x + threadIdx.x;     // [N*OD*HW)
    int pix = idx & (HW - 1);
    int od  = (idx >> 12) & (OD - 1);
    int n   = idx >> 19;
    int o   = od >> 4;
    float acc = 0.0f;
    #pragma unroll
    for (int i = 0; i < NUM_IN; ++i) {
        float rv = r[(((size_t)i * NB + n) * NUM_OUT + o) * HW + pix];
        float uv = uhat[(((size_t)i * NB + n) * OD + od) * HW + pix];
        acc = fmaf(rv, uv, acc);
    }
    p[idx] = acc;
}

// in-place squash over D=16 (v lives in d_out, layout [N][O][D][HW])
__global__ void k_squash(float* __restrict__ v)
{
    int idx = blockIdx.x * blockDim.x + threadIdx.x;     // [N*O*HW)
    int pix = idx & (HW - 1);
    int o   = (idx >> 12) & (NUM_OUT - 1);
    int n   = idx >> 15;
    size_t base = (((size_t)n * OD) + o * OUT_DIM) * HW + pix;
    float s = 0.0f;
    #pragma unroll
    for (int d = 0; d < OUT_DIM; ++d) {
        float x = v[base + (size_t)d * HW];
        s = fmaf(x, x, s);
    }
    float scale = (s / (1.0f + s)) * rsqrtf(s + 1e-9f);
    #pragma unroll
    for (int d = 0; d < OUT_DIM; ++d)
        v[base + (size_t)d * HW] *= scale;
}

// b[i][n][o][pix] += sum_d uhat[i][n][o*16+d][pix] * v[n][o*16+d][pix]
__global__ void k_agree(const float* __restrict__ uhat, const float* __restrict__ v,
                        float* __restrict__ b)
{
    int idx = blockIdx.x * blockDim.x + threadIdx.x;     // [I*N*O*HW)
    int pix = idx & (HW - 1);
    int o   = (idx >> 12) & (NUM_OUT - 1);
    int n   = (idx >> 15) & (NB - 1);
    int i   = idx >> 18;
    size_t ub = (((size_t)i * NB + n) * OD + o * OUT_DIM) * HW + pix;
    size_t vb = (((size_t)n * OD) + o * OUT_DIM) * HW + pix;
    float acc = 0.0f;
    #pragma unroll
    for (int d = 0; d < OUT_DIM; ++d)
        acc = fmaf(uhat[ub + (size_t)d * HW], v[vb + (size_t)d * HW], acc);
    b[idx] += acc;
}

// =====================================================================
extern "C" void kernel_launch(void* const* d_in, const int* in_sizes, int n_in,
                              void* d_out, int out_size, void* d_ws, size_t ws_size,
                              hipStream_t stream)
{
    (void)in_sizes; (void)n_in; (void)out_size; (void)ws_size;
    const float* u = (const float*)d_in[0];   // [N, I, 16, 64, 64]
    const float* W = (const float*)d_in[1];   // [I, 128, 16, 5, 5]
    float* v = (float*)d_out;                 // [N, O, D, H, W]

    // workspace carve-out (floats)
    float* uhat = (float*)d_ws;                                   // I*N*128*HW
    float* b    = uhat + (size_t)NUM_IN * NB * OD * HW;           // I*N*O*HW
    float* c    = b    + (size_t)NUM_IN * NB * NUM_OUT * HW;      // I*N*O*HW (also r)
    float* t0   = c    + (size_t)NUM_IN * NB * NUM_OUT * HW;      // I*N*HW
    float* t1   = t0   + (size_t)NUM_IN * NB * HW;
    float* t2   = t1   + (size_t)NUM_IN * NB * HW;

    const int nB   = NUM_IN * NB * NUM_OUT * HW;   // 2,097,152
    const int nPl  = NUM_IN * NB * HW;             //   262,144
    const int nP   = NB * OD * HW;                 // 4,194,304
    const int nSq  = NB * NUM_OUT * HW;            //   262,144

    // 1) conv -> u_hat  (WMMA implicit GEMM)
    caps_conv_wmma<<<NUM_IN * NB * 128, 256, 0, stream>>>(u, W, uhat);

    // 2) b = 0
    k_zero<<<nB / 256, 256, 0, stream>>>(b, nB);

    // 3) routing
    for (int it = 0; it < 3; ++it) {
        k_max_o<<<nPl / 256, 256, 0, stream>>>(b, t0);
        k_hmax <<<nPl / 256, 256, 0, stream>>>(t0, t1);
        k_vmax <<<nPl / 256, 256, 0, stream>>>(t1, t2);
        k_exp  <<<nPl / 256, 256, 0, stream>>>(b, t2, c, t0);
        k_hsum <<<nPl / 256, 256, 0, stream>>>(t0, t1);
        k_vsum <<<nPl / 256, 256, 0, stream>>>(t1, t2);
        k_div  <<<nB  / 256, 256, 0, stream>>>(c, t2);
        k_p    <<<nP  / 256, 256, 0, stream>>>(c, uhat, v);
        k_squash<<<nSq / 256, 256, 0, stream>>>(v);
        if (it < 2)
            k_agree<<<nB / 256, 256, 0, stream>>>(uhat, v, b);
    }
}